// PatchConv_66812511257256
// MI455X (gfx1250) — compile-verified
//
#include <hip/hip_runtime.h>

typedef __attribute__((ext_vector_type(2))) float v2f;
typedef __attribute__((ext_vector_type(8))) float v8f;

#define KTOT 768   // C*P*P = 3*16*16, contiguous per patch
#define WAVES_PER_BLOCK 8
#define PATCHES_PER_WAVE 16

// out[p] = dot(x[p][0..767], W[0..767]) + bias, computed on the WMMA unit.
// One wave handles 16 patches: A = 16x4 fp32 tile of x, B = 4x16 with every
// column equal to the current W chunk, so every column of D holds the 16
// dot products. Accumulate over 192 chunks of K=4.
__global__ __launch_bounds__(256) void patchconv_wmma_kernel(
    const float* __restrict__ x, const float* __restrict__ w,
    const float* __restrict__ bias, float* __restrict__ out, int npatch)
{
    __shared__ float sW[KTOT];
    for (int i = threadIdx.x; i < KTOT; i += 256) sW[i] = w[i];
    __syncthreads();

    const int lane = threadIdx.x & 31;
    const int wave = threadIdx.x >> 5;
    const int half = lane >> 4;      // 0: K pair {0,1}, 1: K pair {2,3} of each chunk
    const int m    = lane & 15;      // A-matrix row owned by this lane
    const long patch0 =
        ((long)blockIdx.x * WAVES_PER_BLOCK + wave) * PATCHES_PER_WAVE;
    if (patch0 + PATCHES_PER_WAVE > npatch) return;  // wave-uniform; EXEC stays full

    // A layout (16x4 f32): lanes 0-15 hold K={0,1}, lanes 16-31 hold K={2,3}
    const float* xrow = x + (patch0 + m) * (long)KTOT + 2 * half;
    const float* wrow = sW + 2 * half;   // B broadcast: same chunk for all columns

    v8f c = {};
#pragma unroll 8
    for (int k = 0; k < KTOT; k += 4) {
        v2f a = *(const v2f*)(xrow + k);   // global_load_b64, fully streamed
        v2f b = *(const v2f*)(wrow + k);   // ds_load_b64, LDS broadcast
        // 8 args: (neg_a, A, neg_b, B, c_mod, C, reuse_a, reuse_b)
        c = __builtin_amdgcn_wmma_f32_16x16x4_f32(
            false, a, false, b, (short)0, c, false, false);
    }

    const float bv = bias[0];
    // C/D layout: lane 0 column holds D[0..7][0] in c[0..7]; lane 16 holds D[8..15][0].
    if (m == 0) {
        float* dst = out + patch0 + half * 8;
        float4 o0 = make_float4(c[0] + bv, c[1] + bv, c[2] + bv, c[3] + bv);
        float4 o1 = make_float4(c[4] + bv, c[5] + bv, c[6] + bv, c[7] + bv);
        *(float4*)(dst)     = o0;   // global_store_b128
        *(float4*)(dst + 4) = o1;
    }
}

extern "C" void kernel_launch(void* const* d_in, const int* in_sizes, int n_in,
                              void* d_out, int out_size, void* d_ws, size_t ws_size,
                              hipStream_t stream) {
    const float* x    = (const float*)d_in[0];  // [B, N, C, P, P] = [50176, 768]
    const float* w    = (const float*)d_in[1];  // [768]
    const float* bias = (const float*)d_in[2];  // [1]
    float* out = (float*)d_out;                 // [50176]

    const int npatch = out_size;                              // 256*196 = 50176
    const int waves  = npatch / PATCHES_PER_WAVE;             // 3136
    const int blocks = (waves + WAVES_PER_BLOCK - 1) / WAVES_PER_BLOCK;  // 392

    patchconv_wmma_kernel<<<blocks, 256, 0, stream>>>(x, w, bias, out, npatch);
}